// MOT_Symmetric_Potential_Alg_Efficient_81750407512071
// MI455X (gfx1250) — compile-verified
//
#include <hip/hip_runtime.h>
#include <math.h>

typedef __attribute__((ext_vector_type(2))) float v2f;
typedef __attribute__((ext_vector_type(8))) float v8f;

#define BN   8192
#define KD   512
#define EPSR 0.1f
#define SHIFT_C 100.0f

#define TK     128         // K-chunk staged in LDS
#define APITCH 132         // 128 + 4 floats pad -> conflict-free ds_load_b64

// ---------------------------------------------------------------------------
// init: zero the double accumulator (ws is poisoned with 0xAA by the harness)
// ---------------------------------------------------------------------------
__global__ void init_accum_kernel(double* S) { *S = 0.0; }

// ---------------------------------------------------------------------------
// layer 1: h1 = relu(x @ W1 + b1), K=3 contraction -> plain VALU
// ---------------------------------------------------------------------------
__global__ void layer1_kernel(const float* __restrict__ x,
                              const float* __restrict__ W1,
                              const float* __restrict__ b1,
                              float* __restrict__ h) {
  int tid = blockIdx.x * blockDim.x + threadIdx.x;   // BN*KD threads
  int i = tid >> 9;
  int k = tid & 511;
  float acc = b1[k];
  acc = fmaf(x[i * 3 + 0], W1[0 * KD + k], acc);
  acc = fmaf(x[i * 3 + 1], W1[1 * KD + k], acc);
  acc = fmaf(x[i * 3 + 2], W1[2 * KD + k], acc);
  h[tid] = fmaxf(acc, 0.0f);
}

// ---------------------------------------------------------------------------
// fused GEMM + bias + relu: out[8192,512] = relu(A[8192,512] @ W[512,512] + b)
// fp32 WMMA 16x16x4.
//   block = 256 threads = 8 waves -> 64(M) x 128(N) tile
//   A strip (64 x TK) staged in LDS, shared by all 8 waves (8x traffic cut,
//   contiguous b128 global copy). Each wave: 64(M) x 16(N), 4 accumulators,
//   B-fragment reused 4x from registers.
//
// A-frag (16x4 f32, 2 VGPR): lanes 0-15 -> M=lane,    K = {k, k+1}
//                            lanes16-31 -> M=lane-16, K = {k+2, k+3}
// B-frag (4x16 f32, 2 VGPR): lanes 0-15 -> N=lane,    rows K = {k, k+1}
//                            lanes16-31 -> N=lane-16, rows K = {k+2, k+3}
// C/D   (16x16 f32, 8 VGPR): elem r -> M = r + 8*(lane>>4), N = lane&15
// ---------------------------------------------------------------------------
__global__ void gemm_relu_wmma_kernel(const float* __restrict__ A,
                                      const float* __restrict__ W,
                                      const float* __restrict__ bias,
                                      float* __restrict__ out) {
  __shared__ float sh[64 * APITCH];                 // 33 KB

  int tid  = threadIdx.x;
  int lane = tid & 31;
  int hl   = lane >> 4;
  int l15  = lane & 15;
  int w    = tid >> 5;                              // wave in block 0..7

  int m0 = (blockIdx.x >> 2) * 64;                  // 128 m-strips
  int n0 = (blockIdx.x & 3) * 128 + w * 16;         // 4 n-blocks x 8 waves

  v8f c0 = {}, c1 = {}, c2 = {}, c3 = {};
  const float* wp = W + n0 + l15;

  for (int k0 = 0; k0 < KD; k0 += TK) {
    __syncthreads();
    // stage A[m0:m0+64, k0:k0+TK] -> LDS (vectorized, coalesced)
#pragma unroll
    for (int it = 0; it < (64 * TK / 4) / 256; ++it) {
      int idx = it * 256 + tid;                     // vec4 index
      int row = idx >> 5;                           // TK/4 = 32 vec4 per row
      int c4  = (idx & 31) * 4;
      float4 v = *(const float4*)(A + (size_t)(m0 + row) * KD + k0 + c4);
      *(float4*)(&sh[row * APITCH + c4]) = v;
    }
    __syncthreads();

#pragma unroll 2
    for (int kk = 0; kk < TK; kk += 4) {
      int ko = kk + 2 * hl;                         // K offset for half-wave
      v2f b;
      b.x = wp[(size_t)(k0 + ko + 0) * KD];
      b.y = wp[(size_t)(k0 + ko + 1) * KD];
      v2f a0 = *(const v2f*)(&sh[(l15 + 0)  * APITCH + ko]);
      v2f a1 = *(const v2f*)(&sh[(l15 + 16) * APITCH + ko]);
      v2f a2 = *(const v2f*)(&sh[(l15 + 32) * APITCH + ko]);
      v2f a3 = *(const v2f*)(&sh[(l15 + 48) * APITCH + ko]);
      c0 = __builtin_amdgcn_wmma_f32_16x16x4_f32(false, a0, false, b, (short)0, c0, false, false);
      c1 = __builtin_amdgcn_wmma_f32_16x16x4_f32(false, a1, false, b, (short)0, c1, false, false);
      c2 = __builtin_amdgcn_wmma_f32_16x16x4_f32(false, a2, false, b, (short)0, c2, false, false);
      c3 = __builtin_amdgcn_wmma_f32_16x16x4_f32(false, a3, false, b, (short)0, c3, false, false);
    }
  }

  float bn = bias[n0 + l15];
#pragma unroll
  for (int r = 0; r < 8; ++r) {
    int row = m0 + r + 8 * hl;
    out[(size_t)(row + 0)  * KD + n0 + l15] = fmaxf(c0[r] + bn, 0.0f);
    out[(size_t)(row + 16) * KD + n0 + l15] = fmaxf(c1[r] + bn, 0.0f);
    out[(size_t)(row + 32) * KD + n0 + l15] = fmaxf(c2[r] + bn, 0.0f);
    out[(size_t)(row + 48) * KD + n0 + l15] = fmaxf(c3[r] + bn, 0.0f);
  }
}

// ---------------------------------------------------------------------------
// layer 4: phi[i] = tanh(dot(h3[i,:], W4) + b4) * 5 ; one wave per row
// ---------------------------------------------------------------------------
__global__ void phi_kernel(const float* __restrict__ h,
                           const float* __restrict__ W4,
                           const float* __restrict__ b4,
                           float* __restrict__ phi) {
  int wave = (blockIdx.x * blockDim.x + threadIdx.x) >> 5;  // row
  int lane = threadIdx.x & 31;
  const float* row = h + (size_t)wave * KD;
  float s = 0.0f;
#pragma unroll
  for (int k = lane; k < KD; k += 32) s = fmaf(row[k], W4[k], s);
#pragma unroll
  for (int off = 16; off > 0; off >>= 1) s += __shfl_xor(s, off, 32);
  if (lane == 0) phi[wave] = tanhf(s + b4[0]) * 5.0f;
}

// ---------------------------------------------------------------------------
// pairwise: S += sum_{i,j} exp((phi_i + phi_j - 1/(||xi-xj||+1e-9))/eps - C)
// grid (32, 32): blockIdx.x -> 256-row i-tile, blockIdx.y -> 256-col j-tile
// ---------------------------------------------------------------------------
__global__ void pairwise_kernel(const float* __restrict__ x,
                                const float* __restrict__ phi,
                                double* __restrict__ S) {
  __shared__ float xs0[256], xs1[256], xs2[256], ps[256];
  __shared__ double partial[8];

  int i = blockIdx.x * 256 + threadIdx.x;
  int j = blockIdx.y * 256 + threadIdx.x;

  float xi0 = x[i * 3 + 0], xi1 = x[i * 3 + 1], xi2 = x[i * 3 + 2];
  float pi = phi[i];

  xs0[threadIdx.x] = x[j * 3 + 0];
  xs1[threadIdx.x] = x[j * 3 + 1];
  xs2[threadIdx.x] = x[j * 3 + 2];
  ps[threadIdx.x]  = phi[j];
  __syncthreads();

  float s = 0.0f;
#pragma unroll 4
  for (int jj = 0; jj < 256; ++jj) {
    float dx = xi0 - xs0[jj];
    float dy = xi1 - xs1[jj];
    float dz = xi2 - xs2[jj];
    float d2 = fmaf(dx, dx, fmaf(dy, dy, dz * dz));
    float cost = __builtin_amdgcn_rcpf(__builtin_amdgcn_sqrtf(d2) + 1e-9f);
    float alpha = (pi + ps[jj] - cost) * (1.0f / EPSR);
    s += __expf(alpha - SHIFT_C);   // i==j -> alpha ~ -1e10 -> exp underflows to 0
  }

#pragma unroll
  for (int off = 16; off > 0; off >>= 1) s += __shfl_xor(s, off, 32);
  if ((threadIdx.x & 31) == 0) partial[threadIdx.x >> 5] = (double)s;
  __syncthreads();
  if (threadIdx.x == 0) {
    double t = 0.0;
#pragma unroll
    for (int w = 0; w < 8; ++w) t += partial[w];
    atomicAdd(S, t);   // global_atomic_add_f64
  }
}

// ---------------------------------------------------------------------------
// finalize: out = N_e * mean(phi) - eps * (log(S / B^2) + C)
// ---------------------------------------------------------------------------
__global__ void finalize_kernel(const float* __restrict__ phi,
                                const double* __restrict__ S,
                                float* __restrict__ out) {
  __shared__ float red[256];
  float s = 0.0f;
  for (int t = threadIdx.x; t < BN; t += 256) s += phi[t];
  red[threadIdx.x] = s;
  __syncthreads();
  for (int st = 128; st > 0; st >>= 1) {
    if (threadIdx.x < st) red[threadIdx.x] += red[threadIdx.x + st];
    __syncthreads();
  }
  if (threadIdx.x == 0) {
    double mean_phi = (double)red[0] / (double)BN;
    double log_e = log(*S / ((double)BN * (double)BN)) + (double)SHIFT_C;
    out[0] = (float)(2.0 * mean_phi - (double)EPSR * log_e);
  }
}

// ---------------------------------------------------------------------------
extern "C" void kernel_launch(void* const* d_in, const int* in_sizes, int n_in,
                              void* d_out, int out_size, void* d_ws, size_t ws_size,
                              hipStream_t stream) {
  const float* x  = (const float*)d_in[0];
  const float* W1 = (const float*)d_in[1];
  const float* b1 = (const float*)d_in[2];
  const float* W2 = (const float*)d_in[3];
  const float* b2 = (const float*)d_in[4];
  const float* W3 = (const float*)d_in[5];
  const float* b3 = (const float*)d_in[6];
  const float* W4 = (const float*)d_in[7];
  const float* b4 = (const float*)d_in[8];

  float* hA  = (float*)d_ws;                 // 8192*512 f32 (h1, later h3)
  float* hB  = hA + (size_t)BN * KD;         // 8192*512 f32 (h2)
  float* phi = hB + (size_t)BN * KD;         // 8192 f32
  double* S  = (double*)(phi + BN);          // offset is 8-byte aligned

  init_accum_kernel<<<1, 1, 0, stream>>>(S);
  layer1_kernel<<<(BN * KD) / 256, 256, 0, stream>>>(x, W1, b1, hA);
  gemm_relu_wmma_kernel<<<512, 256, 0, stream>>>(hA, W2, b2, hB);   // h2
  gemm_relu_wmma_kernel<<<512, 256, 0, stream>>>(hB, W3, b3, hA);   // h3
  phi_kernel<<<BN / 8, 256, 0, stream>>>(hA, W4, b4, phi);
  pairwise_kernel<<<dim3(32, 32), 256, 0, stream>>>(x, phi, S);
  finalize_kernel<<<1, 256, 0, stream>>>(phi, S, (float*)d_out);
}